// Model_2851858285196
// MI455X (gfx1250) — compile-verified
//
#include <hip/hip_runtime.h>
#include <hip/hip_bf16.h>

typedef __attribute__((ext_vector_type(2))) float v2f;
typedef __attribute__((ext_vector_type(8))) float v8f;

#define NQ      8192
#define ND      32768
#define DK      512
#define TOPK    20
#define THRESH  5e-5f

#define TILE_M  16
#define WAVES   8
#define NT      4                       // 16-wide N tiles per wave
#define TILE_N  (WAVES * NT * 16)       // 512 db rows per iteration
#define LDST    516                     // padded LDS row stride (floats), 4-float pad

// merge-area layout (floats), overlaid on the logits tile after final barrier
#define MERGE_SC 0
#define MERGE_IX (TILE_M * 16 * TOPK)          // 5120
#define MERGE_M  (2 * TILE_M * 16 * TOPK)      // 10240
#define MERGE_S  (2 * TILE_M * 16 * TOPK + 256)
#define SMEM_FLOATS (2 * TILE_M * 16 * TOPK + 512)   // 10752 floats (>16*LDST=8256)

// Insert (v, idx) into a descending-sorted register-resident top-K list using a
// fully-unrolled compare-shift network (constant indices -> stays in VGPRs).
__device__ __forceinline__ void topk_insert(float (&tv)[TOPK], int (&ti)[TOPK],
                                            float v, int idx) {
    if (v <= tv[TOPK - 1]) return;
    #pragma unroll
    for (int p = TOPK - 1; p > 0; --p) {
        bool cp = v > tv[p];
        bool cm = v > tv[p - 1];
        float nv = cp ? (cm ? tv[p - 1] : v) : tv[p];
        int   ni = cp ? (cm ? ti[p - 1] : idx) : ti[p];
        tv[p] = nv; ti[p] = ni;
    }
    bool c0 = v > tv[0];
    ti[0] = c0 ? idx : ti[0];
    tv[0] = c0 ? v : tv[0];
}

__global__ __launch_bounds__(256) void retrieve_topk_kernel(
    const float* __restrict__ Q, const float* __restrict__ D,
    float* __restrict__ out) {
    __shared__ float smem[SMEM_FLOATS];

    const int tid   = threadIdx.x;
    const int wave  = tid >> 5;
    const int lane  = tid & 31;
    const int l16   = lane & 15;
    const int half  = lane >> 4;
    const int qbase = blockIdx.x * TILE_M;

    // A fragment base: lane = query row in tile, K offset 2*half (ISA 16x4 f32 layout)
    const float* aPtr = Q + (size_t)(qbase + l16) * DK + 2 * half;

    // Phase-R role: row r of the tile, column group g (32 cols each)
    const int r = tid & 15;
    const int g = tid >> 4;

    float runM = -__builtin_inff();
    float runS = 0.0f;
    float tv[TOPK]; int ti[TOPK];
    #pragma unroll
    for (int j = 0; j < TOPK; ++j) { tv[j] = -__builtin_inff(); ti[j] = 0; }

    for (int it = 0; it < ND / TILE_N; ++it) {
        const int colBase = it * TILE_N;

        // ---------------- Phase W: WMMA logits, 16 x 64 per wave ----------------
        v8f acc0 = {}, acc1 = {}, acc2 = {}, acc3 = {};
        const int n0 = colBase + wave * (NT * 16) + l16;
        const float* b0 = D + (size_t)(n0 +  0) * DK + 2 * half;
        const float* b1 = D + (size_t)(n0 + 16) * DK + 2 * half;
        const float* b2 = D + (size_t)(n0 + 32) * DK + 2 * half;
        const float* b3 = D + (size_t)(n0 + 48) * DK + 2 * half;
        #pragma unroll 8
        for (int kb = 0; kb < DK; kb += 4) {
            v2f a   = *(const v2f*)(aPtr + kb);
            v2f vb0 = *(const v2f*)(b0 + kb);
            v2f vb1 = *(const v2f*)(b1 + kb);
            v2f vb2 = *(const v2f*)(b2 + kb);
            v2f vb3 = *(const v2f*)(b3 + kb);
            acc0 = __builtin_amdgcn_wmma_f32_16x16x4_f32(false, a, false, vb0, (short)0, acc0, false, false);
            acc1 = __builtin_amdgcn_wmma_f32_16x16x4_f32(false, a, false, vb1, (short)0, acc1, false, false);
            acc2 = __builtin_amdgcn_wmma_f32_16x16x4_f32(false, a, false, vb2, (short)0, acc2, false, false);
            acc3 = __builtin_amdgcn_wmma_f32_16x16x4_f32(false, a, false, vb3, (short)0, acc3, false, false);
        }
        // C/D layout: VGPR rr -> row half*8+rr, lane -> column l16
        {
            const int cb = wave * (NT * 16) + l16;
            #pragma unroll
            for (int rr = 0; rr < 8; ++rr) {
                const int row = half * 8 + rr;
                smem[row * LDST + cb +  0] = acc0[rr];
                smem[row * LDST + cb + 16] = acc1[rr];
                smem[row * LDST + cb + 32] = acc2[rr];
                smem[row * LDST + cb + 48] = acc3[rr];
            }
        }
        __syncthreads();

        // ---------- Phase R: online softmax + top-20 over 32 columns ----------
        #pragma unroll 4
        for (int j = 0; j < 32; ++j) {
            const int c = g * 32 + j;
            float v = smem[r * LDST + c];
            if (v > runM) { runS = runS * __expf(runM - v) + 1.0f; runM = v; }
            else          { runS += __expf(v - runM); }
            topk_insert(tv, ti, v, colBase + c);
        }
        __syncthreads();
    }

    // -------- dump per-thread partial state to LDS (overlays the tile) --------
    {
        const int base = (r * 16 + g) * TOPK;
        #pragma unroll
        for (int j = 0; j < TOPK; ++j) {
            smem[MERGE_SC + base + j] = tv[j];
            smem[MERGE_IX + base + j] = __int_as_float(ti[j]);
        }
        smem[MERGE_M + tid] = runM;
        smem[MERGE_S + tid] = runS;
    }
    __syncthreads();

    // -------- per-row merge: 16 leader threads, exact softmax + threshold --------
    if (tid < TILE_M) {
        const int row = tid;
        float M = -__builtin_inff();
        for (int gg = 0; gg < 16; ++gg)
            M = fmaxf(M, smem[MERGE_M + gg * 16 + row]);
        float S = 0.0f;
        for (int gg = 0; gg < 16; ++gg)
            S += smem[MERGE_S + gg * 16 + row] * __expf(smem[MERGE_M + gg * 16 + row] - M);

        float ftv[TOPK]; int fti[TOPK];
        #pragma unroll
        for (int j = 0; j < TOPK; ++j) { ftv[j] = -__builtin_inff(); fti[j] = 0; }
        for (int gg = 0; gg < 16; ++gg) {
            const int base = (row * 16 + gg) * TOPK;
            for (int j = 0; j < TOPK; ++j) {
                float v = smem[MERGE_SC + base + j];
                if (v <= ftv[TOPK - 1]) break;   // group list is sorted desc
                topk_insert(ftv, fti, v, __float_as_int(smem[MERGE_IX + base + j]));
            }
        }

        const int q = qbase + row;
        float* oScore = out;
        float* oInd   = out + (size_t)NQ * TOPK;
        float* oN     = out + 2 * (size_t)NQ * TOPK;
        float* oMask  = out + 2 * (size_t)NQ * TOPK + NQ;
        const float invS = 1.0f / S;
        int cnt = 0;
        #pragma unroll
        for (int j = 0; j < TOPK; ++j) {
            float sc = __expf(ftv[j] - M) * invS;
            bool keep = sc > THRESH;
            oScore[(size_t)q * TOPK + j] = keep ? sc : 0.0f;
            oInd  [(size_t)q * TOPK + j] = (float)fti[j];
            oMask [(size_t)q * TOPK + j] = keep ? 1.0f : 0.0f;
            cnt += keep ? 1 : 0;
        }
        oN[q] = (float)cnt;
    }
}

extern "C" void kernel_launch(void* const* d_in, const int* in_sizes, int n_in,
                              void* d_out, int out_size, void* d_ws, size_t ws_size,
                              hipStream_t stream) {
    const float* Q = (const float*)d_in[0];
    const float* D = (const float*)d_in[1];
    // d_in[2] is k == 20 (compile-time constant TOPK here)
    float* out = (float*)d_out;
    dim3 grid(NQ / TILE_M);   // 512 workgroups, each owns 16 queries
    dim3 block(256);          // 8 wave32s
    retrieve_topk_kernel<<<grid, block, 0, stream>>>(Q, D, out);
}